// CombinedLoss_3418793967775
// MI455X (gfx1250) — compile-verified
//
#include <hip/hip_runtime.h>
#include <math.h>

// CombinedLoss: total = CE + Dice + 0.1*Boundary over [8,1,256,256] f32.
// out = {total, ce, dice, boundary}
#define BB 8
#define HH 256
#define WW 256
#define NROWS (BB * HH)          // 2048
#define NTOT  (BB * HH * WW)     // 524288

typedef __attribute__((ext_vector_type(2))) float v2f;
typedef __attribute__((ext_vector_type(8))) float v8f;

// ---------------------------------------------------------------------------
// Wave-level (32-lane) sum using the full-precision f32 WMMA:
//   A = 16x4 tile, one (val, 0) pair per lane; B = ones(4x16).
//   D[m][j] = rowsum_m replicated across all 16 columns, so
//   sum over every lane's 8 D registers across the wave = 16 * sum(A).
// Layout-robust: only relies on A having 64 distinct slots and D being the
// full 16x16 matrix spread over the wave. Exact fp32 (mults are by 1.0).
// ---------------------------------------------------------------------------
__device__ __forceinline__ float wave_sum_wmma(float v) {
    v2f a = {v, 0.0f};
    v2f b = {1.0f, 1.0f};
    v8f c = {};
    c = __builtin_amdgcn_wmma_f32_16x16x4_f32(
        /*neg_a=*/false, a, /*neg_b=*/false, b,
        /*c_mod=*/(short)0, c, /*reuse_a=*/false, /*reuse_b=*/false);
    float s = c[0] + c[1] + c[2] + c[3] + c[4] + c[5] + c[6] + c[7];
    s += __shfl_xor(s, 1, 32);
    s += __shfl_xor(s, 2, 32);
    s += __shfl_xor(s, 4, 32);
    s += __shfl_xor(s, 8, 32);
    s += __shfl_xor(s, 16, 32);
    return s * 0.0625f;   // /16 is exact
}

// Deterministic 256-thread block reduction (8 waves). Result in all threads.
__device__ __forceinline__ float block_sum_256(float v, float* swave) {
    float ws = wave_sum_wmma(v);
    int lane = threadIdx.x & 31;
    int wv   = threadIdx.x >> 5;
    __syncthreads();                 // protect swave reuse across calls
    if (lane == 0) swave[wv] = ws;
    __syncthreads();
    return swave[0] + swave[1] + swave[2] + swave[3] +
           swave[4] + swave[5] + swave[6] + swave[7];
}

// ---------------------------------------------------------------------------
// Kernel 1: vertical pass of the exact EDT. One thread per (b, w) column.
// Matches the reference scans: init INF=H+W, d = fg ? 0 : d+1, both ways,
// g = min(fwd, bwd); stores g^2. Also records per-image nonempty flag.
// ---------------------------------------------------------------------------
__global__ void edt_vertical_kernel(const float* __restrict__ target,
                                    float* __restrict__ g2buf,
                                    int* __restrict__ flags) {
    int tid = blockIdx.x * blockDim.x + threadIdx.x;   // 0 .. B*W-1
    int b = tid / WW;
    int w = tid % WW;
    int base = b * HH * WW + w;
    const float INF = (float)(HH + WW);                // 512, as in reference

    float d = INF;
    bool any = false;
    for (int h = 0; h < HH; ++h) {                     // top-down
        bool fg = target[base + h * WW] > 0.5f;
        d = fg ? 0.0f : d + 1.0f;
        any |= fg;
        g2buf[base + h * WW] = d;                      // stash forward dist
    }
    d = INF;
    for (int h = HH - 1; h >= 0; --h) {                // bottom-up
        bool fg = target[base + h * WW] > 0.5f;
        d = fg ? 0.0f : d + 1.0f;
        float g = fminf(g2buf[base + h * WW], d);
        g2buf[base + h * WW] = g * g;                  // store g^2 (exact: int < 2^24)
    }
    if (any) flags[b] = 1;
}

// ---------------------------------------------------------------------------
// Kernel 2: one block (256 threads) per row. Exact row EDT via brute-force
// min_k ((j-k)^2 + g2[k]) from LDS (broadcast reads), fused with the
// elementwise BCE / sigmoid / Dice / boundary accumulators. Per-block
// partial sums written deterministically (no atomics).
// ---------------------------------------------------------------------------
__global__ void row_loss_kernel(const float* __restrict__ logits,
                                const float* __restrict__ target,
                                const float* __restrict__ g2buf,
                                const int* __restrict__ flags,
                                float* __restrict__ partials) {
    __shared__ float sg2[WW];
    __shared__ float swave[8];

    int row = blockIdx.x;            // b*HH + h
    int b   = row / HH;
    int j   = threadIdx.x;           // 0..255 == column
    int idx = row * WW + j;

    sg2[j] = g2buf[idx];
    __syncthreads();

    // exact squared EDT for this pixel
    float jf = (float)j;
    float m = 3.0e38f;
#pragma unroll 8
    for (int k = 0; k < WW; ++k) {
        float diff = jf - (float)k;
        m = fminf(m, __fmaf_rn(diff, diff, sg2[k]));
    }
    float dt = sqrtf(m);
    if (flags[b] == 0) dt = 0.0f;    // empty-mask branch of the reference

    float x = logits[idx];
    float t = fminf(fmaxf(target[idx], 0.0f), 1.0f);

    // stable logaddexp(0,x) - x*t
    float ce = fmaxf(x, 0.0f) + log1pf(expf(-fabsf(x))) - x * t;
    float p  = 1.0f / (1.0f + expf(-x));
    p = fminf(fmaxf(p, 1e-6f), 1.0f - 1e-6f);

    float vals[5] = { ce, p, t, p * t, p * dt };
#pragma unroll
    for (int q = 0; q < 5; ++q) {
        float r = block_sum_256(vals[q], swave);
        if (threadIdx.x == 0) partials[q * NROWS + row] = r;
    }
}

// ---------------------------------------------------------------------------
// Kernel 3: single block folds 2048 partials per quantity (WMMA reduction
// again) and emits {total, ce, dice, boundary}.
// ---------------------------------------------------------------------------
__global__ void finalize_kernel(const float* __restrict__ partials,
                                float* __restrict__ out) {
    __shared__ float swave[8];
    float sums[5];
#pragma unroll
    for (int q = 0; q < 5; ++q) {
        float v = 0.0f;
        for (int i = threadIdx.x; i < NROWS; i += 256)
            v += partials[q * NROWS + i];
        sums[q] = block_sum_256(v, swave);
    }
    if (threadIdx.x == 0) {
        const float S = 1e-6f;
        const float invN = 1.0f / (float)NTOT;
        float ce       = sums[0] * invN;
        float psum     = sums[1];
        float tsum     = sums[2];
        float inter    = sums[3];
        float boundary = sums[4] * invN;
        float dice  = 1.0f - (2.0f * inter + S) / (psum + tsum + S);
        float total = ce + dice + 0.1f * boundary;
        out[0] = total;
        out[1] = ce;
        out[2] = dice;
        out[3] = boundary;
    }
}

// ---------------------------------------------------------------------------
extern "C" void kernel_launch(void* const* d_in, const int* in_sizes, int n_in,
                              void* d_out, int out_size, void* d_ws, size_t ws_size,
                              hipStream_t stream) {
    const float* logits = (const float*)d_in[0];
    const float* target = (const float*)d_in[1];

    // workspace layout: g2 [NTOT] | partials [5*NROWS] | flags [BB]
    float* g2buf    = (float*)d_ws;
    float* partials = g2buf + NTOT;
    int*   flags    = (int*)(partials + 5 * NROWS);

    hipMemsetAsync(flags, 0, BB * sizeof(int), stream);   // capture-safe reset

    edt_vertical_kernel<<<(BB * WW) / 256, 256, 0, stream>>>(target, g2buf, flags);
    row_loss_kernel<<<NROWS, WW, 0, stream>>>(logits, target, g2buf, flags, partials);
    finalize_kernel<<<1, 256, 0, stream>>>(partials, (float*)d_out);
}